// DiscMaker2_53274774339935
// MI455X (gfx1250) — compile-verified
//
#include <hip/hip_runtime.h>
#include <hip/hip_bf16.h>

#define LOG2E 1.44269504088896340736f

typedef __attribute__((ext_vector_type(2))) float v2f;
typedef __attribute__((ext_vector_type(8))) float v8f;

__device__ __forceinline__ v8f wmma4(v2f a, v2f b, v8f c) {
  // V_WMMA_F32_16X16X4_F32 : D = A(16x4) * B(4x16) + C
  return __builtin_amdgcn_wmma_f32_16x16x4_f32(false, a, false, b, (short)0, c, false, false);
}

__device__ __forceinline__ float fexp2(float x) { return __builtin_amdgcn_exp2f(x); }
__device__ __forceinline__ float frcp(float x)  { return __builtin_amdgcn_rcpf(x); }
__device__ __forceinline__ float ftanh(float x) {
  x = fminf(fmaxf(x, -10.f), 10.f);
  float e = fexp2(x * (2.f * LOG2E));
  return (e - 1.f) * frcp(e + 1.f);
}
__device__ __forceinline__ float fsig(float x) {
  x = fminf(fmaxf(x, -30.f), 30.f);
  return frcp(1.f + fexp2(-x * LOG2E));
}

// problem sizes
#define BN   128
#define SN   1024
#define CN   16
#define DN   64
#define HN   65

// LDS strides (padded vs 64 banks)
#define NS_ST  1028
#define S_ST   68
#define H_ST   72
#define INP_ST 72
#define GI_ST  212
#define DEC_ST 68

__global__ __launch_bounds__(512, 1)
void scan_kernel(const float* __restrict__ x, const float* __restrict__ y,
                 const float* __restrict__ s0, const float* __restrict__ init_error,
                 const float* __restrict__ init_gate,
                 const float* __restrict__ W_in, const float* __restrict__ W_state,
                 const float* __restrict__ b_state, const float* __restrict__ traj,
                 const float* __restrict__ W_ih, const float* __restrict__ W_hh,
                 const float* __restrict__ b_ih, const float* __restrict__ b_hh,
                 const float* __restrict__ Wd1, const float* __restrict__ bd1,
                 const float* __restrict__ Wd2, const float* __restrict__ bd2,
                 float* __restrict__ o_out, float* __restrict__ thetaG,
                 float* __restrict__ errG)
{
  __shared__ __align__(16) float ns_l[16 * NS_ST];   // tanh(new_state) 16 x 1024
  __shared__ __align__(16) float s_l[16 * S_ST];     // carried s
  __shared__ __align__(16) float h_l[16 * H_ST];     // carried h (65, K-padded with zeros)
  __shared__ __align__(16) float inp_l[16 * INP_ST]; // [encoded(64), err, 0..]
  __shared__ __align__(16) float gi_l[16 * GI_ST];
  __shared__ __align__(16) float gh_l[16 * GI_ST];
  __shared__ __align__(16) float dec1_l[16 * DEC_ST];
  __shared__ __align__(16) float trajF[4 * 16 * 64]; // B-frags for s@traj^T
  __shared__ __align__(16) float wd1F[4 * 16 * 64];  // B-frags for Wd1^T
  __shared__ __align__(16) float wd2F[16 * 64];      // B-frags for Wd2^T
  __shared__ float Win_l[1024], bst_l[1024];
  __shared__ float bih_l[208], bhh_l[208], bd1_l[64], bd2_l[16], tsq_l[64];
  __shared__ float gate_l[256], theta_l[16], ssq_l[16], xs_l[16], ys_l[16];

  const int tid = threadIdx.x;
  const int w  = tid >> 5;        // wave 0..15
  const int l  = tid & 31;        // lane
  const int m  = l & 15;          // M/N index within frag
  const int hi = l >> 4;          // half select
  const int bt = blockIdx.x;      // batch tile (16 rows)

  // ---------------- persistent register-resident weight fragments ----------------
  // W_state reshaped as B[k=d][n=c*64+e]; wave w owns N-tiles 4w..4w+3
  v2f wst[64];
#pragma unroll
  for (int j = 0; j < 4; ++j) {
    int n = (4 * w + j) * 16 + m;
    int c = n >> 6, e = n & 63;
    const float* base = W_state + c * 4096 + e;
#pragma unroll
    for (int kf = 0; kf < 16; ++kf) {
      int k = 4 * kf + 2 * hi;
      v2f v; v.x = base[k * 64]; v.y = base[(k + 1) * 64];
      wst[j * 16 + kf] = v;
    }
  }
  // GRU weights: B[k][n] = W[n][k], K=65 zero-padded to 68, 13 N-tiles (waves 0..12)
  v2f wih[17], whh[17];
  {
    int n = 16 * w + m;
    bool valid = (w < 13) && (n < 195);
    const float* rih = W_ih + (valid ? n * 65 : 0);
    const float* rhh = W_hh + (valid ? n * 65 : 0);
#pragma unroll
    for (int kf = 0; kf < 17; ++kf) {
      int k = 4 * kf + 2 * hi;
      v2f a, b;
      a.x = (valid && k     < 65) ? rih[k]     : 0.f;
      a.y = (valid && k + 1 < 65) ? rih[k + 1] : 0.f;
      b.x = (valid && k     < 65) ? rhh[k]     : 0.f;
      b.y = (valid && k + 1 < 65) ? rhh[k + 1] : 0.f;
      wih[kf] = a; whh[kf] = b;
    }
  }

  // ---------------- LDS fragment buffers for small weights ----------------
  for (int idx = tid; idx < 4 * 16 * 32; idx += 512) {
    int g = idx >> 9, kf = (idx >> 5) & 15, ll = idx & 31;
    int mm = ll & 15, hh = ll >> 4;
    int k = 4 * kf + 2 * hh;
    int tI = g * 16 + mm;
    trajF[idx * 2]     = traj[tI * 64 + k];      // B[k][t] = traj[t][k]
    trajF[idx * 2 + 1] = traj[tI * 64 + k + 1];
    int n = g * 16 + mm;
    wd1F[idx * 2]      = Wd1[n * 64 + k];        // B[k][n] = Wd1[n][k]
    wd1F[idx * 2 + 1]  = Wd1[n * 64 + k + 1];
  }
  for (int idx = tid; idx < 16 * 32; idx += 512) {
    int kf = idx >> 5, ll = idx & 31;
    int mm = ll & 15, hh = ll >> 4;
    int k = 4 * kf + 2 * hh;
    wd2F[idx * 2]     = Wd2[mm * 64 + k];
    wd2F[idx * 2 + 1] = Wd2[mm * 64 + k + 1];
  }

  // ---------------- other one-time init ----------------
  for (int i = tid; i < 1024; i += 512) { Win_l[i] = W_in[i]; bst_l[i] = b_state[i]; }
  for (int i = tid; i < 208; i += 512) {
    bih_l[i] = (i < 195) ? b_ih[i] : 0.f;
    bhh_l[i] = (i < 195) ? b_hh[i] : 0.f;
  }
  if (tid < 64) bd1_l[tid] = bd1[tid];
  if (tid < 16) bd2_l[tid] = bd2[tid];
  if (tid < 64) { float q = 0.f; for (int d = 0; d < 64; ++d) { float v = traj[tid * 64 + d]; q += v * v; } tsq_l[tid] = q; }
  for (int i = tid; i < 16 * S_ST; i += 512) { int d = i % S_ST; s_l[i] = (d < 64) ? s0[d] : 0.f; }
  for (int i = tid; i < 16 * H_ST; i += 512) h_l[i] = 0.f;
  for (int i = tid; i < 16 * INP_ST; i += 512) { int b = i / INP_ST, j = i % INP_ST; inp_l[i] = (j == 64) ? init_error[b] : 0.f; }
  if (tid < 16) {
    float mx = -1e30f;
    for (int c2 = 0; c2 < 16; ++c2) mx = fmaxf(mx, init_gate[tid * 16 + c2]);
    float sum = 0.f;
    for (int c2 = 0; c2 < 16; ++c2) sum += fexp2((init_gate[tid * 16 + c2] - mx) * LOG2E);
    float r = frcp(sum);
    for (int c2 = 0; c2 < 16; ++c2) gate_l[tid * 16 + c2] = fexp2((init_gate[tid * 16 + c2] - mx) * LOG2E) * r;
    float q = 0.f;
    for (int d = 0; d < 64; ++d) { float v = s0[d]; q += v * v; }
    ssq_l[tid] = q;
    int gb = bt * 16 + tid;
    xs_l[tid] = x[gb * SN];
    ys_l[tid] = y[gb * SN];
  }

  // ================= sequential scan =================
  for (int t = 0; t < SN; ++t) {
    __syncthreads(); // s, ssq, xs/ys, inp[.,64]=err, h, gate ready

    // ---- Phase A: new_state GEMM (+tanh), gh GEMM, s@traj^T (encoded) ----
    {
      v8f accP0 = {}, accP1 = {}, accP2 = {}, accP3 = {};
#pragma unroll
      for (int kf = 0; kf < 16; ++kf) {
        v2f a = *(const v2f*)&s_l[m * S_ST + 4 * kf + 2 * hi];
        accP0 = wmma4(a, wst[0 * 16 + kf], accP0);
        accP1 = wmma4(a, wst[1 * 16 + kf], accP1);
        accP2 = wmma4(a, wst[2 * 16 + kf], accP2);
        accP3 = wmma4(a, wst[3 * 16 + kf], accP3);
      }
      v8f accs[4] = {accP0, accP1, accP2, accP3};
#pragma unroll
      for (int j = 0; j < 4; ++j) {
        int n = (4 * w + j) * 16 + m;
        float wi = Win_l[n], bs = bst_l[n];
#pragma unroll
        for (int v = 0; v < 8; ++v) {
          int row = v + 8 * hi;
          ns_l[row * NS_ST + n] = ftanh(accs[j][v] + xs_l[row] * wi + bs);
        }
      }
      if (w < 13) { // gh = h @ W_hh^T + b_hh
        v8f acc = {};
#pragma unroll
        for (int kf = 0; kf < 17; ++kf) {
          v2f a = *(const v2f*)&h_l[m * H_ST + 4 * kf + 2 * hi];
          acc = wmma4(a, whh[kf], acc);
        }
        int n = 16 * w + m;
        float bb = bhh_l[n];
#pragma unroll
        for (int v = 0; v < 8; ++v) gh_l[(v + 8 * hi) * GI_ST + n] = acc[v] + bb;
      }
      if (w >= 12) { // G = s @ traj^T ; encoded = exp(-g*(||s||^2 - 2G + ||traj||^2))
        int g = w - 12;
        v8f acc = {};
#pragma unroll
        for (int kf = 0; kf < 16; ++kf) {
          v2f a = *(const v2f*)&s_l[m * S_ST + 4 * kf + 2 * hi];
          v2f b = *(const v2f*)&trajF[((g * 16 + kf) * 32 + l) * 2];
          acc = wmma4(a, b, acc);
        }
        int tt = g * 16 + m;
        float tq = tsq_l[tt];
#pragma unroll
        for (int v = 0; v < 8; ++v) {
          int row = v + 8 * hi;
          float d2 = ssq_l[row] - 2.f * acc[v] + tq;
          inp_l[row * INP_ST + tt] = fexp2(-0.5f * LOG2E * d2); // GAMMA=0.5
        }
      }
    }
    __syncthreads();

    // ---- Phase B: gi = inp @ W_ih^T + b_ih ----
    if (w < 13) {
      v8f acc = {};
#pragma unroll
      for (int kf = 0; kf < 17; ++kf) {
        v2f a = *(const v2f*)&inp_l[m * INP_ST + 4 * kf + 2 * hi];
        acc = wmma4(a, wih[kf], acc);
      }
      int n = 16 * w + m;
      float bb = bih_l[n];
#pragma unroll
      for (int v = 0; v < 8; ++v) gi_l[(v + 8 * hi) * GI_ST + n] = acc[v] + bb;
    }
    __syncthreads();

    // ---- Phase C: GRU elementwise, h updated in place ----
    for (int idx = tid; idx < 16 * 65; idx += 512) {
      int b = idx / 65, j = idx - b * 65;
      float gir = gi_l[b * GI_ST + j];
      float giz = gi_l[b * GI_ST + 65 + j];
      float gin = gi_l[b * GI_ST + 130 + j];
      float ghr = gh_l[b * GI_ST + j];
      float ghz = gh_l[b * GI_ST + 65 + j];
      float ghn = gh_l[b * GI_ST + 130 + j];
      float r = fsig(gir + ghr);
      float z = fsig(giz + ghz);
      float nn = ftanh(gin + r * ghn);
      float hn = (1.f - z) * nn + z * h_l[b * H_ST + j];
      h_l[b * H_ST + j] = hn;
      if (j == 64) theta_l[b] = fsig(hn);
    }
    __syncthreads();

    // ---- Phase D1: dec1 = relu(h[:, :64] @ Wd1^T + bd1) ----
    if (w < 4) {
      v8f acc;
      float bb = bd1_l[16 * w + m];
#pragma unroll
      for (int v = 0; v < 8; ++v) acc[v] = bb;
#pragma unroll
      for (int kf = 0; kf < 16; ++kf) {
        v2f a = *(const v2f*)&h_l[m * H_ST + 4 * kf + 2 * hi];
        v2f b = *(const v2f*)&wd1F[((w * 16 + kf) * 32 + l) * 2];
        acc = wmma4(a, b, acc);
      }
#pragma unroll
      for (int v = 0; v < 8; ++v) dec1_l[(v + 8 * hi) * DEC_ST + 16 * w + m] = fmaxf(acc[v], 0.f);
    }
    __syncthreads();

    // ---- Phase D2 (wave 0): dec2 + softmax + gate blend ----
    if (w == 0) {
      v8f acc;
      float bb = bd2_l[m];
#pragma unroll
      for (int v = 0; v < 8; ++v) acc[v] = bb;
#pragma unroll
      for (int kf = 0; kf < 16; ++kf) {
        v2f a = *(const v2f*)&dec1_l[m * DEC_ST + 4 * kf + 2 * hi];
        v2f b = *(const v2f*)&wd2F[(kf * 32 + l) * 2];
        acc = wmma4(a, b, acc);
      }
#pragma unroll
      for (int v = 0; v < 8; ++v) {
        float dv = acc[v];
        float mx = dv;
        mx = fmaxf(mx, __shfl_xor(mx, 1, 32));
        mx = fmaxf(mx, __shfl_xor(mx, 2, 32));
        mx = fmaxf(mx, __shfl_xor(mx, 4, 32));
        mx = fmaxf(mx, __shfl_xor(mx, 8, 32));
        float e = fexp2((dv - mx) * LOG2E);
        float sm = e;
        sm += __shfl_xor(sm, 1, 32);
        sm += __shfl_xor(sm, 2, 32);
        sm += __shfl_xor(sm, 4, 32);
        sm += __shfl_xor(sm, 8, 32);
        float g = e * frcp(sm);
        int row = v + 8 * hi;
        float th = theta_l[row];
        gate_l[row * 16 + m] = g * th + gate_l[row * 16 + m] * (1.f - th);
      }
    }
    __syncthreads();

    // ---- Phase E (wave w -> row w): s_new, ||s||^2, pred/err, outputs, stage x/y ----
    {
      int b = w, gb = bt * 16 + b;
      float g0 = 0.f, g1 = 0.f;
#pragma unroll
      for (int c2 = 0; c2 < 16; ++c2) {
        float gg = gate_l[b * 16 + c2];
        g0 += gg * ns_l[b * NS_ST + c2 * 64 + l];
        g1 += gg * ns_l[b * NS_ST + c2 * 64 + 32 + l];
      }
      s_l[b * S_ST + l] = g0;
      s_l[b * S_ST + 32 + l] = g1;
      float sq = g0 * g0 + g1 * g1;
      sq += __shfl_xor(sq, 1, 32);
      sq += __shfl_xor(sq, 2, 32);
      sq += __shfl_xor(sq, 4, 32);
      sq += __shfl_xor(sq, 8, 32);
      sq += __shfl_xor(sq, 16, 32);
      if (l == 0) ssq_l[b] = sq;
      if (l == 31) { // d = 63
        float pred = fminf(fmaxf(g1, 0.f), 1.f); // relu(x)-relu(x-1)
        float e = pred - ys_l[b];
        inp_l[b * INP_ST + 64] = e;
        o_out[gb * SN + t] = pred;
        errG[t * BN + gb] = e;
        thetaG[t * BN + gb] = theta_l[b];
      }
      if (l == 0 && t + 1 < SN) {
        xs_l[b] = x[gb * SN + t + 1];
        ys_l[b] = y[gb * SN + t + 1];
      }
    }
  }
}

// penalty[i, t, j] = (1 - err[t,j]^2) * theta[t,i]   -> out shape (B, S, B)
__global__ __launch_bounds__(256)
void penalty_kernel(const float* __restrict__ thetaG, const float* __restrict__ errG,
                    float* __restrict__ out2)
{
  unsigned g = blockIdx.x * 256u + threadIdx.x;
  unsigned j = g & 127u;
  unsigned t = (g >> 7) & 1023u;
  unsigned i = g >> 17;
  float e = errG[t * BN + j];
  float th = thetaG[t * BN + i];
  out2[(size_t)i * (SN * BN) + t * BN + j] = (1.f - e * e) * th;
}

extern "C" void kernel_launch(void* const* d_in, const int* in_sizes, int n_in,
                              void* d_out, int out_size, void* d_ws, size_t ws_size,
                              hipStream_t stream) {
  const float* x          = (const float*)d_in[0];
  const float* y          = (const float*)d_in[1];
  const float* s0         = (const float*)d_in[2];
  const float* init_error = (const float*)d_in[3];
  const float* init_gate  = (const float*)d_in[4];
  const float* W_in       = (const float*)d_in[5];
  const float* W_state    = (const float*)d_in[6];
  const float* b_state    = (const float*)d_in[7];
  const float* traj       = (const float*)d_in[8];
  const float* W_ih       = (const float*)d_in[9];
  const float* W_hh       = (const float*)d_in[10];
  const float* b_ih       = (const float*)d_in[11];
  const float* b_hh       = (const float*)d_in[12];
  const float* Wd1        = (const float*)d_in[13];
  const float* bd1        = (const float*)d_in[14];
  const float* Wd2        = (const float*)d_in[15];
  const float* bd2        = (const float*)d_in[16];

  float* o_out  = (float*)d_out;                       // (B, S)
  float* out2   = (float*)d_out + (size_t)BN * SN;     // (B, S, B)
  float* thetaG = (float*)d_ws;                        // (S, B)
  float* errG   = thetaG + (size_t)SN * BN;            // (S, B)

  scan_kernel<<<BN / 16, 512, 0, stream>>>(
      x, y, s0, init_error, init_gate, W_in, W_state, b_state, traj,
      W_ih, W_hh, b_ih, b_hh, Wd1, bd1, Wd2, bd2, o_out, thetaG, errG);

  penalty_kernel<<<(BN * SN * BN) / 256, 256, 0, stream>>>(thetaG, errG, out2);
}